// EfficientPWL_11184094839110
// MI455X (gfx1250) — compile-verified
//
#include <hip/hip_runtime.h>
#include <hip/hip_bf16.h>

#ifndef __has_builtin
#define __has_builtin(x) 0
#endif

// Problem constants (fixed by the reference)
#define PWL_B   32
#define PWL_C   128
#define PWL_T   8192
#define PWL_K   33            // num_cpnts
#define PWL_NS  34            // slopes per channel (K+1)
#define PWL_DT  0.0625f       // 2/(K-1) == 1/16, exact in fp32
#define PWL_INVDT 16.0f

typedef float v4f __attribute__((ext_vector_type(4)));

#if __has_builtin(__builtin_amdgcn_global_load_async_to_lds_b32)
#define PWL_HAVE_ASYNC 1
#else
#define PWL_HAVE_ASYNC 0
#endif

// Evaluate one element against the per-channel LDS table.
// tab[j] = (slope_j, left_breakpoint(ind2(j)), cumbias(ind2(j)), unused)
__device__ __forceinline__ float pwl_eval(float xv, const v4f* __restrict__ tab) {
    // searchsorted(linspace(-1,1,33), x, 'right') for a uniform grid:
    // ind1 = clamp(floor((x+1)/dt) + 1, 0, 33). Breakpoints are exact
    // multiples of 1/16, and fma(x, 16, 16) is exact at those points, so
    // exact-hit semantics match the reference.
    float t = __fmaf_rn(xv, PWL_INVDT, PWL_INVDT);
    int j = __float2int_rd(t) + 1;                 // v_cvt_flr_i32_f32
    j = (j < 0) ? 0 : ((j > PWL_K) ? PWL_K : j);
    v4f e = tab[j];                                // ds_load_b128
    return (xv - e.y) * e.x + e.z;                 // (x - cp)*s + b
}

__global__ __launch_bounds__(256)
void pwl_kernel(const float* __restrict__ x,
                const float* __restrict__ slopes,
                const float* __restrict__ biases,
                float* __restrict__ out)
{
    __shared__ float s_sl[PWL_NS];
    __shared__ v4f   s_tab[PWL_NS];

    const int bc  = blockIdx.x;          // row in [0, B*C)
    const int c   = bc & (PWL_C - 1);    // channel (C=128 is a power of two)
    const int tid = threadIdx.x;

    // ---- Stage A: bring this channel's slope row (34 floats) into LDS ----
#if PWL_HAVE_ASYNC
    if (tid < PWL_NS) {
        // Builtin prototype (from clang diagnostic): AS1 int* src, AS3 int* dst,
        // imm offset, imm cpol.
        __builtin_amdgcn_global_load_async_to_lds_b32(
            (__attribute__((address_space(1))) int*)(slopes + (size_t)c * PWL_NS + tid),
            (__attribute__((address_space(3))) int*)(&s_sl[tid]),
            /*offset=*/0, /*cpol=*/0);
    }
#if __has_builtin(__builtin_amdgcn_s_wait_asynccnt)
    __builtin_amdgcn_s_wait_asynccnt(0);
#else
    asm volatile("s_wait_asynccnt 0" ::: "memory");
#endif
#else
    if (tid < PWL_NS) {
        s_sl[tid] = slopes[(size_t)c * PWL_NS + tid];
    }
#endif
    __syncthreads();

    // ---- Stage B: build the 34-entry (slope, cp, cumbias) table in LDS ----
    // cumbias[k] = biases[c] + dt * sum_{i=1..k} slopes[c][i]   (k in [0,32])
    // Entry j uses ind2 = max(j-1, 0). 34 threads work independently
    // (<=32 fused adds each, once per 8192-element row: negligible).
    if (tid < PWL_NS) {
        int ind2 = tid - 1; if (ind2 < 0) ind2 = 0;
        float acc = 0.0f;
        for (int i = 1; i <= ind2; ++i) acc += s_sl[i] * PWL_DT;
        float cb = acc + biases[c];                      // uniform -> s_load
        float cp = -1.0f + PWL_DT * (float)ind2;         // exact in fp32
        v4f e; e.x = s_sl[tid]; e.y = cp; e.z = cb; e.w = 0.0f;
        s_tab[tid] = e;
    }
    __syncthreads();

    // ---- Stage C: stream the 8192-element row with b128 NT loads/stores ----
    const v4f* __restrict__ xin  = (const v4f*)(x   + (size_t)bc * PWL_T);
    v4f* __restrict__       yout = (v4f*)(out + (size_t)bc * PWL_T);

    // 2048 float4 per row / 256 threads = 8 per thread, fully unrolled so the
    // 8 independent global_load_b128 issue back-to-back and hide HBM latency.
    #pragma unroll
    for (int it = 0; it < (PWL_T / 4) / 256; ++it) {
        const int idx = it * 256 + tid;
        v4f v = __builtin_nontemporal_load(xin + idx);   // single-pass: TH=NT
        v4f r;
        r.x = pwl_eval(v.x, s_tab);
        r.y = pwl_eval(v.y, s_tab);
        r.z = pwl_eval(v.z, s_tab);
        r.w = pwl_eval(v.w, s_tab);
        __builtin_nontemporal_store(r, yout + idx);      // write-once: TH=NT
    }
}

extern "C" void kernel_launch(void* const* d_in, const int* in_sizes, int n_in,
                              void* d_out, int out_size, void* d_ws, size_t ws_size,
                              hipStream_t stream) {
    const float* x      = (const float*)d_in[0];   // [B, C, T] fp32
    const float* slopes = (const float*)d_in[1];   // [C, 34]   fp32
    const float* biases = (const float*)d_in[2];   // [C]       fp32
    float* out = (float*)d_out;                    // [B, C, T] fp32

    (void)in_sizes; (void)n_in; (void)out_size; (void)d_ws; (void)ws_size;

    dim3 grid(PWL_B * PWL_C);   // one block per (b, c) row: 4096 blocks
    dim3 block(256);            // 8 wave32 waves
    hipLaunchKernelGGL(pwl_kernel, grid, block, 0, stream, x, slopes, biases, out);
}